// RotMap_5866925326757
// MI455X (gfx1250) — compile-verified
//
#include <hip/hip_runtime.h>

// D = A(16x4 f32) * B(4x16 f32) + C : per-lane 2 VGPRs for A, 2 for B, 8 for C/D
typedef __attribute__((ext_vector_type(2))) float v2f;
typedef __attribute__((ext_vector_type(8))) float v8f;

// Problem constants (from reference): N=64, D=3, NUM=128, FRAME=512
// x[n,b,c,z,f]  strides (elems): f=1, z=512, c=65536, b=196608, n=589824
// w[a,b,z]      strides (elems): z=1, b=128, a=384
// out[n,a,c,z,f] same strides as x (a in place of b)
#define S_B 196608
#define S_C 65536
#define S_Z 512
#define S_N 589824

__global__ __launch_bounds__(256)
void rotmap_wmma_f32_kernel(const float* __restrict__ x,
                            const float* __restrict__ w,
                            float* __restrict__ out) {
    // blockIdx.x encodes (n, c, z): blk = (n*3 + c)*128 + z
    int blk = blockIdx.x;
    int z = blk & 127;          // NUM = 128
    int t = blk >> 7;
    int c = t % 3;
    int n = t / 3;

    int lane = threadIdx.x & 31;   // wave32
    int wave = threadIdx.x >> 5;   // 8 waves per block
    int row  = lane & 15;
    bool lo  = lane < 16;

    // ---- A matrix: w_z (3x3) padded into 16x4 ----
    // lanes 0-15: VGPR0=K0, VGPR1=K1 ; lanes 16-31: VGPR0=K2, VGPR1=K3(=0)
    float a0 = 0.0f, a1 = 0.0f;
    if (row < 3) {
        int k0 = lo ? 0 : 2;
        a0 = w[row * 384 + k0 * 128 + z];
        if (lo) a1 = w[row * 384 + 128 + z];   // K=1 ; K=3 stays zero
    }
    v2f A;
    A[0] = a0;
    A[1] = a1;

    const size_t base = (size_t)n * S_N + (size_t)c * S_C + (size_t)z * S_Z;
    const float* __restrict__ xb = x + base;
    float* __restrict__ ob = out + base;

    const int bsel = lo ? 0 : 2;   // b-row supplied by this half-wave in VGPR0 of B

    for (int it = 0; it < 4; ++it) {
        int col = it * 128 + wave * 16 + row;   // f column, 0..511

        // ---- B matrix: x rows b=0..2 (b=3 zero), 16 columns ----
        float b0 = xb[(size_t)bsel * S_B + col];   // lanes 0-15: b=0, lanes 16-31: b=2
        float b1 = xb[(size_t)S_B + col];          // b=1 (valid addr for all lanes)
        if (!lo) b1 = 0.0f;                        // K=3 row is zero
        v2f B;
        B[0] = b0;
        B[1] = b1;

        if (it < 3) {
            // warm next 128-column strip through the gfx1250 prefetch path
            __builtin_prefetch(xb + (size_t)bsel * S_B + col + 128, 0, 1);
        }

        v8f Cz = {};
        // 8-arg form: (neg_a, A, neg_b, B, c_mod, C, reuse_a, reuse_b)
        v8f Dv = __builtin_amdgcn_wmma_f32_16x16x4_f32(
            false, A, false, B, (short)0, Cz, false, false);

        // D layout: VGPR r, lanes 0-15 -> M=r, N=lane. Rows a=0..2 are the payload.
        if (lo) {
            ob[0 * S_B + col] = Dv[0];
            ob[1 * S_B + col] = Dv[1];
            ob[2 * S_B + col] = Dv[2];
        }
    }
}

extern "C" void kernel_launch(void* const* d_in, const int* in_sizes, int n_in,
                              void* d_out, int out_size, void* d_ws, size_t ws_size,
                              hipStream_t stream) {
    (void)in_sizes; (void)n_in; (void)d_ws; (void)ws_size; (void)out_size;
    const float* x = (const float*)d_in[0];
    const float* w = (const float*)d_in[1];
    float* out = (float*)d_out;

    // one block per (n, c, z): 64 * 3 * 128 = 24576 blocks, 8 waves each
    dim3 grid(64 * 3 * 128);
    dim3 block(256);
    rotmap_wmma_f32_kernel<<<grid, block, 0, stream>>>(x, w, out);
}